// Transform_6906307412127
// MI455X (gfx1250) — compile-verified
//
#include <hip/hip_runtime.h>

typedef __attribute__((ext_vector_type(2))) float v2f;
typedef __attribute__((ext_vector_type(8))) float v8f;

#define TPS_K 25
#define BS    32
#define IMG_H 384
#define IMG_W 384
#define TPS_EPS 1e-6f
#define LN2F   0.6931471805599453f

// rbf[n,k] = d^2 * log(d + eps), d = L1 distance from pixel grid coord to
// control point k on the 5x5 grid over [-1,1]^2. k >= 25 is zero padding.
// Raw v_log_f32 (log2) + ln2 scale: argument is always in [1e-6, ~4.83],
// so the precise-log denorm/inf guards are unnecessary.
__device__ __forceinline__ float rbf_val(int k, float gx, float gy) {
    const int kk   = (k < TPS_K) ? k : 0;
    const int row  = kk / 5;
    const int col  = kk - row * 5;
    const float cpx = 0.5f * (float)col - 1.0f;
    const float cpy = 0.5f * (float)row - 1.0f;
    const float d  = fabsf(gx - cpx) + fabsf(gy - cpy);
    const float lg = __builtin_amdgcn_logf(d + TPS_EPS);   // v_log_f32 (log2)
    const float v  = (d * d) * (LN2F * lg);
    return (k < TPS_K) ? v : 0.0f;
}

__global__ __launch_bounds__(256) void tps_warp_wmma_kernel(
    const float* __restrict__ frame,   // [BS, H, W, 3]
    const float* __restrict__ theta,   // [BS, 2, 3]
    const float* __restrict__ cpar,    // [BS, 1, 25]
    float* __restrict__ out)           // [BS, H, W, 3]
{
    __shared__ float s_cp[BS * TPS_K]; // 800 floats
    __shared__ float s_th[BS * 6];     // 192 floats

    for (int i = threadIdx.x; i < BS * TPS_K; i += 256) s_cp[i] = cpar[i];
    for (int i = threadIdx.x; i < BS * 6;     i += 256) s_th[i] = theta[i];
    __syncthreads();

    const int lane = threadIdx.x & 31;
    const int wave = threadIdx.x >> 5;
    const int half = lane >> 4;   // 0: lanes 0-15, 1: lanes 16-31
    const int sub  = lane & 15;

    // One wave handles 16 consecutive pixels (same image row) x all 32 batches.
    const int tile  = blockIdx.x * 8 + wave;
    const int pbase = tile * 16;
    const int pidx  = pbase + sub;
    const int oy    = pidx / IMG_W;
    const int ox    = pidx - oy * IMG_W;
    const float gx  = 2.0f * (float)ox / (float)(IMG_W - 1) - 1.0f;
    const float gy  = 2.0f * (float)oy / (float)(IMG_H - 1) - 1.0f;

    // D = A(16 batches x 4 k) * B(4 k x 16 pixels) + C, f32 throughout.
    v8f acc_lo = {0.f,0.f,0.f,0.f,0.f,0.f,0.f,0.f}; // batches 0..15
    v8f acc_hi = {0.f,0.f,0.f,0.f,0.f,0.f,0.f,0.f}; // batches 16..31

    #pragma unroll
    for (int c = 0; c < 7; ++c) {           // K = 25 padded to 28
        const int kA = 4 * c + 2 * half;    // lane's K pair within this slice

        // B operand: rbf values — computed once per (pixel,k) and shared by
        // all 32 batches via the two WMMA chains.
        v2f bm;
        bm.x = rbf_val(kA,     gx, gy);
        bm.y = rbf_val(kA + 1, gx, gy);

        // A operands: control params from LDS (zero-padded past k=24).
        v2f a_lo, a_hi;
        a_lo.x = (kA     < TPS_K) ? s_cp[sub        * TPS_K + kA    ] : 0.0f;
        a_lo.y = (kA + 1 < TPS_K) ? s_cp[sub        * TPS_K + kA + 1] : 0.0f;
        a_hi.x = (kA     < TPS_K) ? s_cp[(sub + 16) * TPS_K + kA    ] : 0.0f;
        a_hi.y = (kA + 1 < TPS_K) ? s_cp[(sub + 16) * TPS_K + kA + 1] : 0.0f;

        acc_lo = __builtin_amdgcn_wmma_f32_16x16x4_f32(
            false, a_lo, false, bm, (short)0, acc_lo, false, false);
        acc_hi = __builtin_amdgcn_wmma_f32_16x16x4_f32(
            false, a_hi, false, bm, (short)0, acc_hi, false, false);
    }

    // Epilogue: lane holds res for its own pixel (pbase+sub) and 16 batches.
    // Adjacent lanes -> adjacent x -> coalesced gathers/stores.
    const int pout = (oy * IMG_W + ox) * 3;

    auto sample_store = [&](int b, float res) {
        const float t0 = s_th[b*6+0], t1 = s_th[b*6+1], t2 = s_th[b*6+2];
        const float t3 = s_th[b*6+3], t4 = s_th[b*6+4], t5 = s_th[b*6+5];
        const float wx = t0 * gx + t1 * gy + t2 + res;
        const float wy = t3 * gx + t4 * gy + t5 + res;
        const float fx = 0.5f * (wx + 1.0f) * (float)(IMG_W - 1);
        const float fy = 0.5f * (wy + 1.0f) * (float)(IMG_H - 1);
        int x0 = (int)floorf(fx);
        int y0 = (int)floorf(fy);
        int x1 = x0 + 1;
        int y1 = y0 + 1;
        x0 = min(max(x0, 0), IMG_W - 1);
        x1 = min(max(x1, 0), IMG_W - 1);
        y0 = min(max(y0, 0), IMG_H - 1);
        y1 = min(max(y1, 0), IMG_H - 1);
        const float x0f = (float)x0, x1f = (float)x1;
        const float y0f = (float)y0, y1f = (float)y1;
        const float wa = (x1f - fx) * (y1f - fy);
        const float wb = (x1f - fx) * (fy - y0f);
        const float wc = (fx - x0f) * (y1f - fy);
        const float wd = (fx - x0f) * (fy - y0f);
        const int ib = b * (IMG_H * IMG_W);
        const float* pa = frame + (ib + y0 * IMG_W + x0) * 3;
        const float* pb = frame + (ib + y1 * IMG_W + x0) * 3;
        const float* pc = frame + (ib + y0 * IMG_W + x1) * 3;
        const float* pd = frame + (ib + y1 * IMG_W + x1) * 3;
        float* po = out + ib * 3 + pout;
        #pragma unroll
        for (int ch = 0; ch < 3; ++ch) {
            po[ch] = wa * pa[ch] + wb * pb[ch] + wc * pc[ch] + wd * pd[ch];
        }
    };

    #pragma unroll
    for (int r = 0; r < 8; ++r) {
        sample_store(8 * half + r,      acc_lo[r]);  // batches 0..15
        sample_store(8 * half + r + 16, acc_hi[r]);  // batches 16..31
    }
}

extern "C" void kernel_launch(void* const* d_in, const int* in_sizes, int n_in,
                              void* d_out, int out_size, void* d_ws, size_t ws_size,
                              hipStream_t stream) {
    (void)in_sizes; (void)n_in; (void)out_size; (void)d_ws; (void)ws_size;
    const float* frame = (const float*)d_in[0];
    const float* theta = (const float*)d_in[1];
    const float* cpar  = (const float*)d_in[2];
    float* out = (float*)d_out;

    const int tiles  = (IMG_H * IMG_W) / 16;  // 9216 pixel tiles
    const int blocks = tiles / 8;             // 8 waves (tiles) per 256-thread block
    tps_warp_wmma_kernel<<<blocks, 256, 0, stream>>>(frame, theta, cpar, out);
}